// LatencyEncoder_72885595013562
// MI455X (gfx1250) — compile-verified
//
#include <hip/hip_runtime.h>

// LatencyEncoder: x(256,4096) f32 -> one-hot spike train (256, 64, 4096) f32.
// Memory-bound: 256 MB of write-once output => ~11us floor at 23.3 TB/s.
// Strategy: 1 thread per 4 columns of one batch row; compute spike time once;
// emit the 64 time-steps as 64 coalesced 128-bit NON-TEMPORAL stores with
// immediate offsets (fully unrolled, zero address math in the loop).
// 2D grid (x = column tile, y = batch row) => no integer division anywhere.

#define T_STEPS 64
#define TAU     10.0f
#define EPS     1e-7f
#define N_COLS  4096              // reference inner dim (compile-time)
#define NVEC    (N_COLS / 4)      // 1024 float4 per row

typedef __attribute__((ext_vector_type(4))) float v4f;

__device__ __forceinline__ int spike_time(float x) {
    // matches reference: sigmoid -> -TAU*log(s+eps) -> clip[0,63] -> floor
    float s   = 1.0f / (1.0f + expf(-x));
    float lat = -TAU * logf(s + EPS);
    lat = fminf(fmaxf(lat, 0.0f), (float)(T_STEPS - 1));
    return (int)lat;  // lat >= 0, so trunc == floor
}

__global__ __launch_bounds__(256) void latency_encode_kernel(
    const float* __restrict__ x,   // (B, N) f32
    float* __restrict__ out)       // (B, T, N) f32
{
    const int nv = blockIdx.x * blockDim.x + threadIdx.x;  // float4 index in row
    const int b  = blockIdx.y;                             // batch row
    if (nv >= NVEC) return;

    const v4f* __restrict__ xin = (const v4f*)x;
    const size_t in_idx = (size_t)b * NVEC + nv;

    // gfx1250 prefetch path (global_prefetch_b8); input is tiny, cost ~0.
    __builtin_prefetch(xin + in_idx, 0, 1);

    v4f v = xin[in_idx];  // global_load_b128, coalesced

    int t0 = spike_time(v.x);
    int t1 = spike_time(v.y);
    int t2 = spike_time(v.z);
    int t3 = spike_time(v.w);

    // base of column slab: out[b, 0, nv*4] — all strides are compile-time
    float* base = out + (size_t)b * (T_STEPS * N_COLS) + (size_t)nv * 4;

    // Fully unrolled: 64 stores at immediate offsets ts*N_COLS*4 bytes
    // (max 63*16384 = 1,032,192 < 2^23, fits the signed 24-bit IOFFSET).
    #pragma unroll
    for (int ts = 0; ts < T_STEPS; ++ts) {
        v4f sp;
        sp.x = (t0 == ts) ? 1.0f : 0.0f;
        sp.y = (t1 == ts) ? 1.0f : 0.0f;
        sp.z = (t2 == ts) ? 1.0f : 0.0f;
        sp.w = (t3 == ts) ? 1.0f : 0.0f;
        // Write-once stream, never re-read: non-temporal to spare the 192MB L2.
        __builtin_nontemporal_store(sp, (v4f*)(base + ts * N_COLS));
    }
}

extern "C" void kernel_launch(void* const* d_in, const int* in_sizes, int n_in,
                              void* d_out, int out_size, void* d_ws, size_t ws_size,
                              hipStream_t stream) {
    const float* x = (const float*)d_in[0];
    float* out     = (float*)d_out;

    const int n_elems = in_sizes[0];         // B * N
    const int n_rows  = n_elems / N_COLS;    // B = 256

    const int block = 256;                   // 8 wave32 per block
    dim3 grid((NVEC + block - 1) / block,    // 4 column tiles per row
              n_rows);                       // 256 rows

    latency_encode_kernel<<<grid, block, 0, stream>>>(x, out);
}